// RGCNEncoder_39152921870698
// MI455X (gfx1250) — compile-verified
//
#include <hip/hip_runtime.h>
#include <hip/hip_bf16.h>

typedef __attribute__((ext_vector_type(2))) float v2f;
typedef __attribute__((ext_vector_type(8))) float v8f;

#define D 128
#define LDS_STRIDE 260   // 256 + 4 pad -> conflict-free A-fragment ds reads

// ---------------- CSR build ----------------

__global__ void k_zero_i32(int* p, int n) {
    int i = blockIdx.x * blockDim.x + threadIdx.x;
    if (i < n) p[i] = 0;
}

__global__ void k_degree(const int* __restrict__ dst, int* __restrict__ deg, int e) {
    int i = blockIdx.x * blockDim.x + threadIdx.x;
    if (i < e) atomicAdd(&deg[dst[i]], 1);
}

// inclusive scan per 256-block; off[i+1] gets local inclusive sum; bsum[b] = block total
__global__ void k_scan_block(const int* __restrict__ deg, int* __restrict__ off,
                             int* __restrict__ bsum, int n) {
    __shared__ int s[256];
    int i = blockIdx.x * 256 + threadIdx.x;
    int v = (i < n) ? deg[i] : 0;
    s[threadIdx.x] = v;
    __syncthreads();
    for (int d = 1; d < 256; d <<= 1) {
        int t = (threadIdx.x >= d) ? s[threadIdx.x - d] : 0;
        __syncthreads();
        s[threadIdx.x] += t;
        __syncthreads();
    }
    if (i < n) off[i + 1] = s[threadIdx.x];
    if (threadIdx.x == 255) bsum[blockIdx.x] = s[255];
}

__global__ void k_scan_sums(int* bsum, int nb) {
    if (blockIdx.x == 0 && threadIdx.x == 0) {
        int acc = 0;
        for (int b = 0; b < nb; ++b) { int t = bsum[b]; bsum[b] = acc; acc += t; }
    }
}

__global__ void k_scan_add(int* __restrict__ off, const int* __restrict__ bsum, int n) {
    int i = blockIdx.x * 256 + threadIdx.x;
    if (i < n) off[i + 1] += bsum[blockIdx.x];
    if (i == 0) off[0] = 0;
}

__global__ void k_fill(const int* __restrict__ src, const int* __restrict__ dst,
                       const int* __restrict__ off, int* __restrict__ cur,
                       int* __restrict__ csr, int e) {
    int i = blockIdx.x * blockDim.x + threadIdx.x;
    if (i < e) {
        int d = dst[i];
        int p = atomicAdd(&cur[d], 1);
        csr[off[d] + p] = src[i];
    }
}

// ---------------- fused SAGE: mean-aggregate (LDS) + dual GEMM via fp32 WMMA ----------------
// out[g, :] = mean_nbr(xsrc)[g] @ Wl^T + bl + xdst[g] @ Wr^T   (optional relu)
// Block: 256 threads (8 waves) handles 16 dst rows x all 128 output cols.
__global__ __launch_bounds__(256) void k_sage(
    const float* __restrict__ xsrc, const float* __restrict__ xdst,
    const int* __restrict__ off, const int* __restrict__ csr,
    const float* __restrict__ Wl, const float* __restrict__ Wr,
    const float* __restrict__ bl, float* __restrict__ out,
    int nDst, int relu)
{
    __shared__ float A2[16 * LDS_STRIDE];   // [row][0:128)=mean, [128:256)=xdst
    const int wave = threadIdx.x >> 5;
    const int lane = threadIdx.x & 31;
    const int mbase = blockIdx.x * 16;
    const int c0 = lane * 4;

    // Phase 1: wave w aggregates rows 2w, 2w+1 (each lane owns 4 consecutive features)
    for (int rr = 0; rr < 2; ++rr) {
        const int r = wave * 2 + rr;
        const int g = mbase + r;
        float* row = &A2[r * LDS_STRIDE];
        if (g < nDst) {
            const int beg = off[g], end = off[g + 1];
            float ax = 0.f, ay = 0.f, az = 0.f, aw = 0.f;
            for (int j = beg; j < end; ++j) {
                const int s = csr[j];
                const float4 v = *reinterpret_cast<const float4*>(xsrc + (size_t)s * D + c0);
                ax += v.x; ay += v.y; az += v.z; aw += v.w;
            }
            const float sc = 1.0f / fmaxf((float)(end - beg), 1.0f);
            row[c0 + 0] = ax * sc; row[c0 + 1] = ay * sc;
            row[c0 + 2] = az * sc; row[c0 + 3] = aw * sc;
            const float4 xv = *reinterpret_cast<const float4*>(xdst + (size_t)g * D + c0);
            row[D + c0 + 0] = xv.x; row[D + c0 + 1] = xv.y;
            row[D + c0 + 2] = xv.z; row[D + c0 + 3] = xv.w;
        } else {
            row[c0 + 0] = 0.f; row[c0 + 1] = 0.f; row[c0 + 2] = 0.f; row[c0 + 3] = 0.f;
            row[D + c0 + 0] = 0.f; row[D + c0 + 1] = 0.f; row[D + c0 + 2] = 0.f; row[D + c0 + 3] = 0.f;
        }
    }
    __syncthreads();

    // Phase 2: wave w computes output cols [16w, 16w+16). fp32 WMMA 16x16x4:
    //  A frag: lanes 0-15 -> M=lane, K={k,k+1}; lanes 16-31 -> M=lane-16, K={k+2,k+3}
    //  B frag: symmetric with N=lane&15
    const int lh = lane & 15;
    const int hi = lane >> 4;                 // 0 or 1
    const int jcol = wave * 16 + lh;          // output column (0..127)
    const float* arow = &A2[lh * LDS_STRIDE + 2 * hi];
    const float* wlrow = Wl + (size_t)jcol * D + 2 * hi;   // Wl[j][k] contiguous in k
    const float* wrrow = Wr + (size_t)jcol * D + 2 * hi;

    v8f c = {};
    #pragma unroll
    for (int k0 = 0; k0 < D; k0 += 4) {
        const v2f a = *reinterpret_cast<const v2f*>(arow + k0);
        const v2f b = *reinterpret_cast<const v2f*>(wlrow + k0);
        c = __builtin_amdgcn_wmma_f32_16x16x4_f32(false, a, false, b, (short)0, c, false, false);
    }
    #pragma unroll
    for (int k0 = 0; k0 < D; k0 += 4) {
        const v2f a = *reinterpret_cast<const v2f*>(arow + D + k0);
        const v2f b = *reinterpret_cast<const v2f*>(wrrow + k0);
        c = __builtin_amdgcn_wmma_f32_16x16x4_f32(false, a, false, b, (short)0, c, false, false);
    }

    const float bb = bl[jcol];
    #pragma unroll
    for (int i = 0; i < 8; ++i) {
        const int gr = mbase + i + 8 * hi;    // C/D layout: VGPR i, lanes>=16 -> M+8
        if (gr < nDst) {
            float v = c[i] + bb;
            if (relu) v = fmaxf(v, 0.0f);
            out[(size_t)gr * D + jcol] = v;
        }
    }
}

// ---------------- launch ----------------

extern "C" void kernel_launch(void* const* d_in, const int* in_sizes, int n_in,
                              void* d_out, int out_size, void* d_ws, size_t ws_size,
                              hipStream_t stream) {
    const float* x_player = (const float*)d_in[0];
    const float* x_team   = (const float*)d_in[1];
    const int*   ei_pt    = (const int*)d_in[2];
    const int*   ei_tp    = (const int*)d_in[3];
    const float* Wl0_pt = (const float*)d_in[4];
    const float* Wr0_pt = (const float*)d_in[5];
    const float* Wl0_tp = (const float*)d_in[6];
    const float* Wr0_tp = (const float*)d_in[7];
    const float* Wl1_pt = (const float*)d_in[8];
    const float* Wr1_pt = (const float*)d_in[9];
    const float* Wl1_tp = (const float*)d_in[10];
    const float* Wr1_tp = (const float*)d_in[11];
    const float* bl0_pt = (const float*)d_in[12];
    const float* bl0_tp = (const float*)d_in[13];
    const float* bl1_pt = (const float*)d_in[14];
    const float* bl1_tp = (const float*)d_in[15];

    const int N = in_sizes[0] / D;       // 100000
    const int E = in_sizes[2] / 2;       // 600000

    const int* src_pt = ei_pt;           // player -> team
    const int* dst_pt = ei_pt + E;
    const int* src_tp = ei_tp;           // team -> player
    const int* dst_tp = ei_tp + E;

    // workspace carve-up (ints, then 256B-aligned floats)
    char* p = (char*)d_ws;
    auto take_i = [&](size_t n) { int* r = (int*)p; p += n * sizeof(int); return r; };
    int* deg_pt = take_i(N);
    int* deg_tp = take_i(N);
    int* cur_pt = take_i(N);
    int* cur_tp = take_i(N);             // deg_pt..cur_tp contiguous: zeroed together
    int* off_pt = take_i(N + 1);
    int* off_tp = take_i(N + 1);
    int* bsum   = take_i(1024);
    int* csr_pt = take_i(E);
    int* csr_tp = take_i(E);
    p = (char*)(((size_t)p + 255) & ~(size_t)255);
    float* h_player = (float*)p; p += (size_t)N * D * sizeof(float);
    float* h_team   = (float*)p; p += (size_t)N * D * sizeof(float);

    float* o_player = (float*)d_out;             // tuple order: (o_player, o_team)
    float* o_team   = (float*)d_out + (size_t)N * D;

    const int TB = 256;
    const int nbN  = (N + TB - 1) / TB;          // scan blocks
    const int nbE  = (E + TB - 1) / TB;
    const int nb4N = (4 * N + TB - 1) / TB;
    const int nbM  = (N + 15) / 16;              // sage blocks

    // --- CSR build (shared by both layers) ---
    k_zero_i32<<<nb4N, TB, 0, stream>>>(deg_pt, 4 * N);
    k_degree<<<nbE, TB, 0, stream>>>(dst_pt, deg_pt, E);
    k_degree<<<nbE, TB, 0, stream>>>(dst_tp, deg_tp, E);

    k_scan_block<<<nbN, TB, 0, stream>>>(deg_pt, off_pt, bsum, N);
    k_scan_sums<<<1, 1, 0, stream>>>(bsum, nbN);
    k_scan_add<<<nbN, TB, 0, stream>>>(off_pt, bsum, N);

    k_scan_block<<<nbN, TB, 0, stream>>>(deg_tp, off_tp, bsum, N);
    k_scan_sums<<<1, 1, 0, stream>>>(bsum, nbN);
    k_scan_add<<<nbN, TB, 0, stream>>>(off_tp, bsum, N);

    k_fill<<<nbE, TB, 0, stream>>>(src_pt, dst_pt, off_pt, cur_pt, csr_pt, E);
    k_fill<<<nbE, TB, 0, stream>>>(src_tp, dst_tp, off_tp, cur_tp, csr_tp, E);

    // --- Layer 0 (ReLU) ---
    k_sage<<<nbM, TB, 0, stream>>>(x_player, x_team,   off_pt, csr_pt, Wl0_pt, Wr0_pt, bl0_pt, h_team,   N, 1);
    k_sage<<<nbM, TB, 0, stream>>>(x_team,   x_player, off_tp, csr_tp, Wl0_tp, Wr0_tp, bl0_tp, h_player, N, 1);

    // --- Layer 1 (no activation) ---
    k_sage<<<nbM, TB, 0, stream>>>(h_player, h_team,   off_pt, csr_pt, Wl1_pt, Wr1_pt, bl1_pt, o_team,   N, 0);
    k_sage<<<nbM, TB, 0, stream>>>(h_team,   h_player, off_tp, csr_tp, Wl1_tp, Wr1_tp, bl1_tp, o_player, N, 0);
}